// BEVPoolV2_79783312491037
// MI455X (gfx1250) — compile-verified
//
#include <hip/hip_runtime.h>
#include <hip/hip_bf16.h>

#define MAXN   40
#define VOX_PER_BLOCK 8           // 8 waves (wave32) per 256-thread block
#define BLOCK_THREADS 256

// ---------------------------------------------------------------------------
// Main kernel: one wave32 per voxel, lane handles 2 channels (C==64).
// Rank index streams are staged into LDS via gfx1250 async global->LDS loads.
// ---------------------------------------------------------------------------
__global__ __launch_bounds__(BLOCK_THREADS)
void bevpool_v2_main(const float* __restrict__ depth,
                     const float* __restrict__ feat,
                     const int*   __restrict__ ranks_depth,
                     const int*   __restrict__ ranks_feat,
                     float*       __restrict__ out,
                     int nDepth, int nFeatRows, int nVox)
{
    __shared__ int s_rd[VOX_PER_BLOCK * MAXN];   // 320 ints
    __shared__ int s_rf[VOX_PER_BLOCK * MAXN];   // 320 ints

    const int tid = threadIdx.x;
    const int p0  = blockIdx.x * (VOX_PER_BLOCK * MAXN);   // base point index

    const unsigned long long rd_base = (unsigned long long)(size_t)ranks_depth;
    const unsigned long long rf_base = (unsigned long long)(size_t)ranks_feat;

    // ---- Stage 320+320 ints: async copy global -> LDS (ASYNCcnt path) ----
    {
        unsigned lds_a = (unsigned)(size_t)(&s_rd[tid]);
        unsigned lds_b = (unsigned)(size_t)(&s_rf[tid]);
        unsigned voff  = (unsigned)((p0 + tid) * 4);
        asm volatile("global_load_async_to_lds_b32 %0, %1, %2"
                     :: "v"(lds_a), "v"(voff), "s"(rd_base) : "memory");
        asm volatile("global_load_async_to_lds_b32 %0, %1, %2"
                     :: "v"(lds_b), "v"(voff), "s"(rf_base) : "memory");
    }
    if (tid < 64) {   // waves 0..1 only: wave-uniform branch, EXEC all-ones inside
        unsigned lds_a = (unsigned)(size_t)(&s_rd[256 + tid]);
        unsigned lds_b = (unsigned)(size_t)(&s_rf[256 + tid]);
        unsigned voff  = (unsigned)((p0 + 256 + tid) * 4);
        asm volatile("global_load_async_to_lds_b32 %0, %1, %2"
                     :: "v"(lds_a), "v"(voff), "s"(rd_base) : "memory");
        asm volatile("global_load_async_to_lds_b32 %0, %1, %2"
                     :: "v"(lds_b), "v"(voff), "s"(rf_base) : "memory");
    }
    // Each wave drains its own async issues, then barrier for cross-wave LDS vis.
    asm volatile("s_wait_asynccnt 0" ::: "memory");
    __syncthreads();

    // ---- Compute: wave w owns voxel v; lane owns channels [2*lane, 2*lane+1] ----
    const int wv   = tid >> 5;
    const int lane = tid & 31;
    const int v    = blockIdx.x * VOX_PER_BLOCK + wv;
    if (v >= nVox) return;

    const float2* __restrict__ feat2 = (const float2*)feat;

    float ax = 0.0f, ay = 0.0f;
    #pragma unroll 4
    for (int k = 0; k < MAXN; ++k) {
        // indices are uniform across the wave -> force SGPR for scalar gather
        int rd = __builtin_amdgcn_readfirstlane(s_rd[wv * MAXN + k]);
        int rf = __builtin_amdgcn_readfirstlane(s_rf[wv * MAXN + k]);

        float dval = 0.0f;
        if (rd < nDepth) dval = depth[rd];             // uniform -> s_load

        float fx = 0.0f, fy = 0.0f;
        if (rf < nFeatRows) {                          // pad row -> zeros
            float2 f = feat2[(size_t)rf * 32 + lane];  // global_load_b64
            fx = f.x; fy = f.y;
        }
        ax = fmaf(dval, fx, ax);
        ay = fmaf(dval, fy, ay);
    }
    float2 res; res.x = ax; res.y = ay;
    ((float2*)out)[(size_t)v * 32 + lane] = res;       // global_store_b64
}

// ---------------------------------------------------------------------------
// Generic fallback (any maxn): one lane per (voxel, channel-pair), indices
// read straight from global (still sequential -> cache friendly).
// ---------------------------------------------------------------------------
__global__ __launch_bounds__(BLOCK_THREADS)
void bevpool_v2_generic(const float* __restrict__ depth,
                        const float* __restrict__ feat,
                        const int*   __restrict__ ranks_depth,
                        const int*   __restrict__ ranks_feat,
                        float*       __restrict__ out,
                        int nDepth, int nFeatRows, int nVox, int maxn)
{
    int t    = blockIdx.x * blockDim.x + threadIdx.x;
    int v    = t >> 5;
    int lane = t & 31;
    if (v >= nVox) return;

    const float2* __restrict__ feat2 = (const float2*)feat;
    float ax = 0.0f, ay = 0.0f;
    long base = (long)v * maxn;
    for (int k = 0; k < maxn; ++k) {
        int rd = ranks_depth[base + k];
        int rf = ranks_feat[base + k];
        float dval = (rd < nDepth) ? depth[rd] : 0.0f;
        float fx = 0.0f, fy = 0.0f;
        if (rf < nFeatRows) {
            float2 f = feat2[(size_t)rf * 32 + lane];
            fx = f.x; fy = f.y;
        }
        ax = fmaf(dval, fx, ax);
        ay = fmaf(dval, fy, ay);
    }
    float2 res; res.x = ax; res.y = ay;
    ((float2*)out)[(size_t)v * 32 + lane] = res;
}

extern "C" void kernel_launch(void* const* d_in, const int* in_sizes, int n_in,
                              void* d_out, int out_size, void* d_ws, size_t ws_size,
                              hipStream_t stream)
{
    (void)n_in; (void)d_ws; (void)ws_size;
    const float* depth       = (const float*)d_in[0];
    const float* feat        = (const float*)d_in[1];
    const int*   ranks_depth = (const int*)d_in[2];
    const int*   ranks_feat  = (const int*)d_in[3];
    float*       out         = (float*)d_out;

    const int C         = 64;                       // fixed by reference
    const int nDepth    = in_sizes[0];              // flat depth rows (B*N*D*IH*IW)
    const int nFeatRows = in_sizes[1] / C;          // feat rows (B*N*IH*IW)
    const int nRanks    = in_sizes[2];
    const int nVox      = out_size / C;             // 40,000
    const int maxn      = (nVox > 0) ? (nRanks / nVox) : 0;

    if (maxn == MAXN && (nVox % VOX_PER_BLOCK) == 0) {
        dim3 grid(nVox / VOX_PER_BLOCK);
        dim3 block(BLOCK_THREADS);
        bevpool_v2_main<<<grid, block, 0, stream>>>(
            depth, feat, ranks_depth, ranks_feat, out, nDepth, nFeatRows, nVox);
    } else {
        int threads = nVox * 32;
        dim3 grid((threads + BLOCK_THREADS - 1) / BLOCK_THREADS);
        dim3 block(BLOCK_THREADS);
        bevpool_v2_generic<<<grid, block, 0, stream>>>(
            depth, feat, ranks_depth, ranks_feat, out, nDepth, nFeatRows, nVox, maxn);
    }
}